// MyModel_87522843559982
// MI455X (gfx1250) — compile-verified
//
#include <hip/hip_runtime.h>
#include <cstdint>

// ---------------- problem constants (from reference) ----------------
#define NRES   262144      // NUM_RESULTS
#define NCLS   91          // NUM_CLASSES
#define NC     90          // classes excluding background
#define MAXB   10          // MAX_BOXES
#define SCORE_THR 0.01f
#define IOU_THR   0.5f

#define CHUNK  4096        // score elements per TDM tile (16 KB)
#define TROWS  128         // rows per transpose block

typedef unsigned int u32;
typedef unsigned long long u64;
typedef u32 v4u  __attribute__((ext_vector_type(4)));
typedef int  v8i  __attribute__((ext_vector_type(8)));
typedef int  v4i  __attribute__((ext_vector_type(4)));

// -------------------------------------------------------------------
// Tensor Data Mover: 1-D tile load Global -> LDS (CDNA5 ISA ch. 8).
// D# group0: count=1 | lds_addr | global_addr[56:0] | type=2
// D# group1: data_size=4B, tensor_dim0=n, tensor_dim1=1, tile_dim0=n,
//            tile_dim1=1 (single row), strides benign.
// 6-arg builtin on this toolchain:
//   (uint32x4 g0, int32x8 g1, int32x4 g2, int32x4 g3, int32x8 gx, i32 cpol)
// Tracked by TENSORcnt; caller (one wave) waits with s_wait_tensorcnt.
// -------------------------------------------------------------------
__device__ __forceinline__ void tdm_load_f32_1d(u32 ldsByteAddr,
                                                const float* gsrc,
                                                int nElems) {
  u64 ga = (u64)(uintptr_t)gsrc;
  v4u g0;
  g0.x = 1u;                                              // count=1, user D#
  g0.y = ldsByteAddr;                                     // lds_addr
  g0.z = (u32)ga;                                         // global_addr[31:0]
  g0.w = (u32)((ga >> 32) & 0x01FFFFFFu) | (2u << 30);    // addr[56:32]|type=2
  v8i g1;
  g1[0] = (2 << 16);                                      // data_size = 4B
  g1[1] = (int)(((u32)nElems & 0xFFFFu) << 16);           // tensor_dim0 lo16
  g1[2] = (int)((((u32)nElems >> 16) & 0xFFFFu)           // tensor_dim0 hi16
                | (1u << 16));                            // tensor_dim1 = 1
  g1[3] = (int)(((u32)nElems & 0xFFFFu) << 16);           // tile_dim0
  g1[4] = 1;                                              // tile_dim1=1, dim2=0
  g1[5] = nElems;                                         // dim0 stride (benign)
  g1[6] = 0;
  g1[7] = 0;
  v4i g2 = (v4i){0, 0, 0, 0};
  v4i g3 = (v4i){0, 0, 0, 0};
  v8i gx = (v8i){0, 0, 0, 0, 0, 0, 0, 0};
  __builtin_amdgcn_tensor_load_to_lds(g0, g1, g2, g3, gx, 0);
}

__device__ __forceinline__ float iou_f(float4 a, float4 b) {
  float yy1 = fmaxf(a.x, b.x), xx1 = fmaxf(a.y, b.y);
  float yy2 = fminf(a.z, b.z), xx2 = fminf(a.w, b.w);
  float inter = fmaxf(yy2 - yy1, 0.0f) * fmaxf(xx2 - xx1, 0.0f);
  float areaA = (a.z - a.x) * (a.w - a.y);
  float areaB = (b.z - b.x) * (b.w - b.y);
  return inter / fmaxf(areaA + areaB - inter, 1e-8f);
}

__device__ __forceinline__ float sigmoid_f(float x) {
  return 1.0f / (1.0f + __expf(-x));   // native v_exp_f32 (TRANS32 co-exec)
}

// ---------------- kernel 1: box decode ----------------
__global__ __launch_bounds__(256)
void decode_kernel(const float4* __restrict__ loc,
                   const float* __restrict__ priors,
                   float4* __restrict__ boxes) {
  int i = blockIdx.x * 256 + threadIdx.x;
  if (i >= NRES) return;
  float4 l  = loc[i];
  float ycp = priors[i];
  float xcp = priors[NRES + i];
  float hp  = priors[2 * NRES + i];
  float wp  = priors[3 * NRES + i];
  // constant divides -> reciprocal multiplies (streaming kernel, keep VALU lean)
  float yc = (l.x * 0.1f) * hp + ycp;
  float xc = (l.y * 0.1f) * wp + xcp;
  float h  = __expf(l.z * 0.2f) * hp;
  float w  = __expf(l.w * 0.2f) * wp;
  boxes[i] = make_float4(yc - 0.5f * h, xc - 0.5f * w,
                         yc + 0.5f * h, xc + 0.5f * w);
}

// ---------------- kernel 2: sigmoid + mask + transpose ----------------
// Produces scoresT[c][i] = masked sigmoid, c = 0..89 (class c+1).
__global__ __launch_bounds__(TROWS)
void score_kernel(const float* __restrict__ logits,
                  float* __restrict__ scoresT) {
  __shared__ float tile[TROWS * NCLS];          // 46.6 KB
  const int i0  = blockIdx.x * TROWS;
  const int tid = threadIdx.x;
  const float* src = logits + (size_t)i0 * NCLS;
  __builtin_prefetch(src + TROWS * NCLS, 0, 1); // global_prefetch_b8
  for (int k = tid; k < TROWS * NCLS; k += TROWS) {
    tile[k] = sigmoid_f(src[k]);
  }
  __syncthreads();
  // stride-91 LDS reads: gcd(91,64)=1 -> bank-conflict free
  for (int c = 1; c < NCLS; ++c) {
    float s = tile[tid * NCLS + c];
    scoresT[(size_t)(c - 1) * NRES + i0 + tid] = (s > SCORE_THR) ? s : -1.0f;
  }
}

// ---------------- kernel 3: per-class greedy NMS ----------------
// One block per class, 1024 threads (32 wave32 waves).
// USE_T: stream the pre-masked score row via TDM into double-buffered LDS,
// overlapping the DMA of chunk j+1 with the VALU scan of chunk j.
template <bool USE_T>
__global__ __launch_bounds__(1024)
void nms_kernel(const float* __restrict__ scoresT,
                const float* __restrict__ logits,
                const float4* __restrict__ boxes,
                float* __restrict__ cand) {
  const int c    = blockIdx.x;       // 0..89  (class id = c+1)
  const int tid  = threadIdx.x;
  const int wid  = tid >> 5;
  const int lane = tid & 31;

  __shared__ float4 selBox[MAXB];
  __shared__ int    nSelSh;
  __shared__ float  redS[32];
  __shared__ int    redI[32];

  if (tid == 0) nSelSh = 0;
  __syncthreads();

  const float* row = scoresT + (size_t)c * NRES;

  for (int r = 0; r < MAXB; ++r) {
    const int nSel = nSelSh;
    float bestS = 0.0f;       // eligible scores are > SCORE_THR > 0
    int   bestI = -1;

    if constexpr (USE_T) {
      __shared__ float sbuf[2][CHUNK];   // 32 KB double buffer
      // prologue: DMA chunk 0, wave 0 waits on TENSORcnt, barrier releases all
      if (wid == 0) {
        tdm_load_f32_1d((u32)(uintptr_t)&sbuf[0][0], row, CHUNK);
        __builtin_amdgcn_s_wait_tensorcnt(0);
      }
      __syncthreads();
      int p = 0;
      const int nChunks = NRES / CHUNK;  // 64
      for (int j = 0; j < nChunks; ++j) {
        if (wid == 0 && j + 1 < nChunks) {
          tdm_load_f32_1d((u32)(uintptr_t)&sbuf[p ^ 1][0],
                          row + (size_t)(j + 1) * CHUNK, CHUNK);
        }
        const int base = j * CHUNK;
        for (int t = tid; t < CHUNK; t += 1024) {
          float s = sbuf[p][t];
          if (s > bestS) {                     // lazy: boxes touched rarely
            int i = base + t;
            float4 b = boxes[i];
            bool ok = true;
            for (int k = 0; k < nSel; ++k)
              if (iou_f(b, selBox[k]) > IOU_THR) { ok = false; break; }
            if (ok) { bestS = s; bestI = i; }
          }
        }
        __syncthreads();                       // all done reading sbuf[p]
        if (wid == 0 && j + 1 < nChunks) __builtin_amdgcn_s_wait_tensorcnt(0);
        __syncthreads();                       // next buffer is ready
        p ^= 1;
      }
    } else {
      // fallback: strided logit column + on-the-fly sigmoid (L2-resident)
      for (int i = tid; i < NRES; i += 1024) {
        float s = sigmoid_f(logits[(size_t)i * NCLS + (c + 1)]);
        if (s > SCORE_THR && s > bestS) {
          float4 b = boxes[i];
          bool ok = true;
          for (int k = 0; k < nSel; ++k)
            if (iou_f(b, selBox[k]) > IOU_THR) { ok = false; break; }
          if (ok) { bestS = s; bestI = i; }
        }
      }
    }

    // wave32 argmax reduction (first-index tie-break like jnp.argmax)
    for (int off = 16; off > 0; off >>= 1) {
      float oS = __shfl_xor(bestS, off, 32);
      int   oI = __shfl_xor(bestI, off, 32);
      if (oS > bestS || (oS == bestS && (u32)oI < (u32)bestI)) {
        bestS = oS; bestI = oI;
      }
    }
    if (lane == 0) { redS[wid] = bestS; redI[wid] = bestI; }
    __syncthreads();
    if (wid == 0) {
      bestS = redS[lane]; bestI = redI[lane];
      for (int off = 16; off > 0; off >>= 1) {
        float oS = __shfl_xor(bestS, off, 32);
        int   oI = __shfl_xor(bestI, off, 32);
        if (oS > bestS || (oS == bestS && (u32)oI < (u32)bestI)) {
          bestS = oS; bestI = oI;
        }
      }
      if (lane == 0) {
        float* cr = cand + ((size_t)c * MAXB + r) * 8;
        if (bestS > 0.0f) {
          float4 bb = boxes[bestI];
          selBox[nSelSh] = bb;                // self-IOU==1 suppresses reuse
          nSelSh = nSelSh + 1;
          cr[0] = bestS;  cr[1] = (float)bestI; cr[2] = (float)(c + 1);
          cr[3] = bb.x;   cr[4] = bb.y;  cr[5] = bb.z;  cr[6] = bb.w;
        } else {
          cr[0] = -1.0f;  cr[1] = 0.0f;  cr[2] = (float)(c + 1);
          cr[3] = 0.0f;   cr[4] = 0.0f;  cr[5] = 0.0f;  cr[6] = 0.0f;
        }
      }
    }
    __syncthreads();   // selBox/nSelSh visible before next round
  }
}

// ---------------- kernel 4: global top-10 over 900 candidates ----------------
__global__ __launch_bounds__(1024)
void topk_kernel(const float* __restrict__ cand, float* __restrict__ out) {
  const int tid  = threadIdx.x;
  const int wid  = tid >> 5;
  const int lane = tid & 31;
  __shared__ float sc[NC * MAXB];   // 900 scores
  __shared__ float redS[32];
  __shared__ int   redI[32];
  if (tid < NC * MAXB) sc[tid] = cand[(size_t)tid * 8];
  __syncthreads();
  for (int r = 0; r < MAXB; ++r) {
    float bestS = -3.4e38f;
    int   bestI = -1;
    if (tid < NC * MAXB) { bestS = sc[tid]; bestI = tid; }
    for (int off = 16; off > 0; off >>= 1) {
      float oS = __shfl_xor(bestS, off, 32);
      int   oI = __shfl_xor(bestI, off, 32);
      if (oS > bestS || (oS == bestS && (u32)oI < (u32)bestI)) {
        bestS = oS; bestI = oI;
      }
    }
    if (lane == 0) { redS[wid] = bestS; redI[wid] = bestI; }
    __syncthreads();
    if (wid == 0) {
      bestS = redS[lane]; bestI = redI[lane];
      for (int off = 16; off > 0; off >>= 1) {
        float oS = __shfl_xor(bestS, off, 32);
        int   oI = __shfl_xor(bestI, off, 32);
        if (oS > bestS || (oS == bestS && (u32)oI < (u32)bestI)) {
          bestS = oS; bestI = oI;
        }
      }
      if (lane == 0) {
        const float* cr = cand + (size_t)bestI * 8;
        float v = (bestS > 0.0f) ? 1.0f : 0.0f;
        for (int k = 0; k < 7; ++k) out[r * 7 + k] = cr[k] * v;
        sc[bestI] = -3.4e38f;
      }
    }
    __syncthreads();
  }
}

// ---------------- host launcher ----------------
extern "C" void kernel_launch(void* const* d_in, const int* in_sizes, int n_in,
                              void* d_out, int out_size, void* d_ws,
                              size_t ws_size, hipStream_t stream) {
  const float* loc    = (const float*)d_in[0];   // [N,4]
  const float* logits = (const float*)d_in[1];   // [N,91]
  const float* priors = (const float*)d_in[2];   // [4,N]
  float* out = (float*)d_out;                    // [10,7]

  char* ws = (char*)d_ws;
  float*  cand    = (float*)ws;                              // 900*8 floats
  float4* boxes   = (float4*)(ws + 32768);                   // 4 MB
  float*  scoresT = (float*)(ws + 32768 + (size_t)NRES * 16);
  size_t need = 32768 + (size_t)NRES * 16 + (size_t)NC * NRES * 4;
  bool useT = (ws_size >= need);

  decode_kernel<<<NRES / 256, 256, 0, stream>>>((const float4*)loc, priors,
                                                boxes);
  if (useT) {
    score_kernel<<<NRES / TROWS, TROWS, 0, stream>>>(logits, scoresT);
    nms_kernel<true><<<NC, 1024, 0, stream>>>(scoresT, logits, boxes, cand);
  } else {
    nms_kernel<false><<<NC, 1024, 0, stream>>>(scoresT, logits, boxes, cand);
  }
  topk_kernel<<<1, 1024, 0, stream>>>(cand, out);
}